// LightplaneSplatter_31353261261020
// MI455X (gfx1250) — compile-verified
//
#include <hip/hip_runtime.h>

typedef __attribute__((ext_vector_type(2))) float v2f;
typedef __attribute__((ext_vector_type(8))) float v8f;

#define NUM_SAMPLES     64
#define NUM_SAMPLES_INF 8
#define GD 128
#define GH 128
#define GW 128
#define GC 16
#define WAVES_PER_BLOCK 8
#define NQUAD_UNI  (NUM_SAMPLES / 4)       // 16 iterations, samples 0..63
#define NQUAD_INF  (NUM_SAMPLES_INF / 4)   // 2 iterations, samples 64..71

// ---------------------------------------------------------------------------
// Zero-init the output grid (harness poisons d_out with 0xAA before timing).
// ---------------------------------------------------------------------------
__global__ void splat_zero_kernel(float4* __restrict__ out, long n4) {
    long i      = (long)blockIdx.x * blockDim.x + threadIdx.x;
    long stride = (long)gridDim.x * blockDim.x;
    for (; i < n4; i += stride)
        out[i] = make_float4(0.0f, 0.0f, 0.0f, 0.0f);
}

// ---------------------------------------------------------------------------
// One wave32 per ray; each iteration processes 4 samples (32 (sample,corner)
// rows, one per lane). Two V_WMMA_F32_16X16X4_F32 outer products
// (w_rows (x) encoding) produce all 512 splat values with the D layout
// (lane = channel, VGPR = row) needed for channel-contiguous atomic scatter:
// each half-wave's 16 global_atomic_add_f32 land in one 64B line.
// Addresses are 32-bit byte offsets (grid is 134MB) -> saddr+voffset atomics.
// ---------------------------------------------------------------------------
__global__ __launch_bounds__(256) void splat_wmma_kernel(
    const float* __restrict__ dirs,  const float* __restrict__ orgs,
    const float* __restrict__ nearv, const float* __restrict__ farv,
    const float* __restrict__ enc,   const int*   __restrict__ gidx,
    float* __restrict__ grid, int N)
{
    const int lane = threadIdx.x & 31;
    const int wave = __builtin_amdgcn_readfirstlane((int)(threadIdx.x >> 5));
    const int ray  = blockIdx.x * WAVES_PER_BLOCK + wave;
    if (ray >= N) return;  // wave-uniform: EXEC stays all-1s for WMMA

    // Wave-uniform ray data -> scalar loads.
    const float ddx = dirs[3 * ray + 0];
    const float ddy = dirs[3 * ray + 1];
    const float ddz = dirs[3 * ray + 2];
    const float ox  = orgs[3 * ray + 0];
    const float oy  = orgs[3 * ray + 1];
    const float oz  = orgs[3 * ray + 2];
    const float nr  = nearv[ray];
    const float fr  = farv[ray];
    const int   b   = gidx[ray];

    // B matrix (4x16 f32): only row K=0 carries the encoding.
    v2f bm;
    bm.x = (lane < 16) ? enc[(size_t)ray * GC + lane] : 0.0f;
    bm.y = 0.0f;

    const bool lo   = (lane < 16);
    const int  cx   = lane & 1;
    const int  cy   = (lane >> 1) & 1;
    const int  cz   = (lane >> 2) & 1;
    const int  sub  = (lane >> 3) & 3;       // which of the 4 samples
    const int  half = (lane & 16) ? 8 : 0;   // D-row group for this half-wave
    const int  ch4  = (lane & 15) << 2;      // channel byte offset

    const float invS  = 1.0f / (float)NUM_SAMPLES;
    const float invSi = 1.0f / (float)NUM_SAMPLES_INF;
    const float dnf   = fr - nr;
    const float invf  = __builtin_amdgcn_rcpf(fr);

    // Process 4 samples given per-lane depth t: weights+indices on all 32
    // lanes, two WMMAs (pair A = lanes 0-15 rows, pair B = lanes 16-31 rows),
    // then 16 coalesced atomic scatters.
    auto process4 = [&](float t) {
        float px = ox + t * ddx, py = oy + t * ddy, pz = oz + t * ddz;
        float vx = (px + 1.0f) * (0.5f * (float)(GW - 1));
        float vy = (py + 1.0f) * (0.5f * (float)(GH - 1));
        float vz = (pz + 1.0f) * (0.5f * (float)(GD - 1));
        float bx = floorf(vx), by = floorf(vy), bz = floorf(vz);
        float fx = vx - bx,    fy = vy - by,    fz = vz - bz;
        int jx = (int)bx + cx, jy = (int)by + cy, jz = (int)bz + cz;

        float wx = cx ? fx : (1.0f - fx);
        float wy = cy ? fy : (1.0f - fy);
        float wz = cz ? fz : (1.0f - fz);
        float w  = wx * wy * wz;

        bool inb = ((unsigned)jx < (unsigned)GW) &
                   ((unsigned)jy < (unsigned)GH) &
                   ((unsigned)jz < (unsigned)GD);
        if (!inb) w = 0.0f;                        // MASK_OOB

        int ixc = min(max(jx, 0), GW - 1);
        int iyc = min(max(jy, 0), GH - 1);
        int izc = min(max(jz, 0), GD - 1);
        int flat = ((b * GD + izc) * GH + iyc) * GW + ixc;
        int offB = flat << 6;                      // byte offset of voxel row

        // Pair A: rows from lanes 0-15.
        v2f a1; a1.x = lo ? w : 0.0f; a1.y = 0.0f;
        // Pair B: rows from lanes 16-31 (one half-swap shuffle).
        float wsw = __shfl_xor(w, 16, 32);
        v2f a2; a2.x = lo ? wsw : 0.0f; a2.y = 0.0f;

        v8f acc1 = {};
        acc1 = __builtin_amdgcn_wmma_f32_16x16x4_f32(
            false, a1, false, bm, (short)0, acc1, false, false);
        v8f acc2 = {};
        acc2 = __builtin_amdgcn_wmma_f32_16x16x4_f32(
            false, a2, false, bm, (short)0, acc2, false, false);

        #pragma unroll
        for (int r = 0; r < 8; ++r) {
            int o = __shfl(offB, r + half, 32);
            atomicAdd((float*)((char*)grid + (unsigned)(o + ch4)), acc1[r]);
        }
        #pragma unroll
        for (int r = 0; r < 8; ++r) {
            int o = __shfl(offB, 16 + r + half, 32);
            atomicAdd((float*)((char*)grid + (unsigned)(o + ch4)), acc2[r]);
        }
    };

    // Uniform-depth samples: s = 4q + sub, all < 64 -> branch-free loop.
    for (int q = 0; q < NQUAD_UNI; ++q) {
        int   s = 4 * q + sub;
        float t = nr + dnf * (((float)s + 0.5f) * invS);
        process4(t);
    }
    // Beyond-far samples, uniform in disparity: s = 64 + 4q + sub.
    for (int q = 0; q < NQUAD_INF; ++q) {
        int   si   = 4 * q + sub;                  // 0..7
        float j    = (float)(si + 1) * invSi;
        float disp = invf + (1e-4f - invf) * j;
        float t    = __builtin_amdgcn_rcpf(disp);
        process4(t);
    }
}

extern "C" void kernel_launch(void* const* d_in, const int* in_sizes, int n_in,
                              void* d_out, int out_size, void* d_ws, size_t ws_size,
                              hipStream_t stream) {
    const float* dirs  = (const float*)d_in[0];
    const float* orgs  = (const float*)d_in[1];
    const float* nearv = (const float*)d_in[2];
    const float* farv  = (const float*)d_in[3];
    const float* enc   = (const float*)d_in[4];
    const int*   gidx  = (const int*)  d_in[5];
    float*       out   = (float*)d_out;

    const int N = in_sizes[2];  // one entry per ray

    // 1) Zero the grid (out_size = 1*128*128*128*16 floats, divisible by 4).
    long n4 = (long)out_size / 4;
    splat_zero_kernel<<<2048, 256, 0, stream>>>((float4*)d_out, n4);

    // 2) Splat: one wave per ray, 8 waves per block.
    int blocks = (N + WAVES_PER_BLOCK - 1) / WAVES_PER_BLOCK;
    splat_wmma_kernel<<<blocks, 256, 0, stream>>>(dirs, orgs, nearv, farv,
                                                  enc, gidx, out, N);
}